// Loss_9062380994628
// MI455X (gfx1250) — compile-verified
//
#include <hip/hip_runtime.h>
#include <stdint.h>
#include <math.h>

// YOLO-v1 loss, S=7, B=2, C=20, channels = 5*B+C = 30.
// Memory-bound streaming reduction: stage 256-cell tiles to LDS with CDNA5
// async global->LDS DMA (b128, NT hint, ASYNCcnt-tracked), compute per-cell
// loss from conflict-free LDS, deterministic two-pass shuffle reduction.

#define SS      7
#define CH      30
#define TPB     256
#define NWAVES  (TPB / 32)
#define TILE_F  (TPB * CH)      // 7680 floats per tensor per tile (30720 B)
#define TILE_V4 (TILE_F / 4)    // 1920 float4s
#define V4ITER  (TILE_V4 / TPB) // 7 full iterations (+ one half-width)

#if defined(__gfx1250__) && __has_builtin(__builtin_amdgcn_global_load_async_to_lds_b128) && __has_builtin(__builtin_amdgcn_s_wait_asynccnt)
#define USE_ASYNC_LDS 1
#endif

#if defined(USE_ASYNC_LDS)
// Builtin expects (V4i __device__*, V4i __shared__*, imm offset, imm cpol),
// where V4i is a *generic* vector (vector_size), not ext_vector_type.
typedef int v4i __attribute__((vector_size(4 * sizeof(int))));
typedef __attribute__((address_space(1))) v4i gas_v4i;
typedef __attribute__((address_space(3))) v4i lds_v4i;
#define CPOL_NT 1   // TH[2:0]=1 (non-temporal): 385MB stream > 192MB L2
#endif

__device__ __forceinline__ float cell_loss(const float* __restrict__ p,
                                           const float* __restrict__ tg) {
  const float invS = 1.0f / 7.0f;

  float objf   = (tg[4] > 0.0f) ? 1.0f : 0.0f;
  float noobjf = 1.0f - objf;

  // no-object confidence loss over conf channels {4, 9}
  float d4 = p[4] - tg[4];
  float d9 = p[9] - tg[9];
  float loss_noobj = noobjf * (d4 * d4 + d9 * d9);

  // class loss over channels [10, 30)
  float loss_class = 0.0f;
#pragma unroll
  for (int k = 10; k < 30; ++k) {
    float d = p[k] - tg[k];
    loss_class += d * d;
  }
  loss_class *= objf;

  // target box 0 -> xyxy
  float tx = tg[0] * invS, ty = tg[1] * invS, tw = tg[2], th = tg[3];
  float t_x0 = tx - 0.5f * tw, t_y0 = ty - 0.5f * th;
  float t_x1 = tx + 0.5f * tw, t_y1 = ty + 0.5f * th;
  float area_t = (t_x1 - t_x0) * (t_y1 - t_y0);

  float iou0 = 0.0f, iou1 = 0.0f;
#pragma unroll
  for (int b = 0; b < 2; ++b) {
    const float* pb = p + 5 * b;
    float px = pb[0] * invS, py = pb[1] * invS, pw = pb[2], ph = pb[3];
    float x0 = px - 0.5f * pw, y0 = py - 0.5f * ph;
    float x1 = px + 0.5f * pw, y1 = py + 0.5f * ph;
    float ltx = fmaxf(x0, t_x0), lty = fmaxf(y0, t_y0);
    float rbx = fminf(x1, t_x1), rby = fminf(y1, t_y1);
    float iw = fmaxf(rbx - ltx, 0.0f), ih = fmaxf(rby - lty, 0.0f);
    float inter = iw * ih;
    float ap = (x1 - x0) * (y1 - y0);
    float un = fmaxf(ap + area_t - inter, 1e-10f);
    float v = inter / un;
    if (b == 0) iou0 = v; else iou1 = v;
  }

  int   bb      = (iou1 > iou0) ? 1 : 0;     // jnp.argmax: first max wins
  float max_iou = fmaxf(iou0, iou1);
  const float* pr = p  + 5 * bb;
  const float* tr = tg + 5 * bb;

  float dx = pr[0] - tr[0], dy = pr[1] - tr[1];
  float loss_xy = dx * dx + dy * dy;
  float dw = sqrtf(pr[2]) - sqrtf(tr[2]);
  float dh = sqrtf(pr[3]) - sqrtf(tr[3]);
  float loss_wh = dw * dw + dh * dh;
  float dc = pr[4] - max_iou;
  float loss_obj = dc * dc;

  return objf * (5.0f * (loss_xy + loss_wh) + loss_obj)
       + 0.5f * loss_noobj + loss_class;
}

// Deterministic wave32 tree sum (fixed combine order).
__device__ __forceinline__ float wave_sum(float v) {
#pragma unroll
  for (int off = 16; off > 0; off >>= 1)
    v += __shfl_down(v, off, 32);
  return v;
}

__global__ __launch_bounds__(TPB) void yolo_loss_partial(
    const float* __restrict__ pred, const float* __restrict__ tgt,
    float* __restrict__ partial, int n_cells) {
  __shared__ __align__(16) float sP[TILE_F];
  __shared__ __align__(16) float sT[TILE_F];
  __shared__ float sW[NWAVES];

  const int tid  = threadIdx.x;
  const int base = blockIdx.x * TPB;

  float cell = 0.0f;
  const bool full = (base + TPB) <= n_cells;   // uniform per block

  if (full) {
    const float* gp = pred + (size_t)base * CH;
    const float* gt = tgt  + (size_t)base * CH;
#if defined(USE_ASYNC_LDS)
    // Coalesced async DMA, 16B granules: lane copies float4 (i*256 + tid)
    // of the tile for both tensors; tail half-iteration covers the last
    // 128 float4s. All transfers ASYNCcnt-tracked, non-temporal.
#pragma unroll
    for (int i = 0; i < V4ITER; ++i) {
      int v = i * TPB + tid;
      __builtin_amdgcn_global_load_async_to_lds_b128(
          (gas_v4i*)(gp + 4 * v), (lds_v4i*)(&sP[4 * v]), 0, CPOL_NT);
      __builtin_amdgcn_global_load_async_to_lds_b128(
          (gas_v4i*)(gt + 4 * v), (lds_v4i*)(&sT[4 * v]), 0, CPOL_NT);
    }
    if (tid < TPB / 2) {
      int v = V4ITER * TPB + tid;
      __builtin_amdgcn_global_load_async_to_lds_b128(
          (gas_v4i*)(gp + 4 * v), (lds_v4i*)(&sP[4 * v]), 0, CPOL_NT);
      __builtin_amdgcn_global_load_async_to_lds_b128(
          (gas_v4i*)(gt + 4 * v), (lds_v4i*)(&sT[4 * v]), 0, CPOL_NT);
    }
    __builtin_amdgcn_s_wait_asynccnt(0);
    __syncthreads();
#else
    // Fallback: sync coalesced staging through VGPRs.
#pragma unroll
    for (int i = 0; i < V4ITER; ++i) {
      int v = i * TPB + tid;
      float4 a = *(const float4*)(gp + 4 * v);
      float4 b = *(const float4*)(gt + 4 * v);
      *(float4*)(&sP[4 * v]) = a;
      *(float4*)(&sT[4 * v]) = b;
    }
    if (tid < TPB / 2) {
      int v = V4ITER * TPB + tid;
      float4 a = *(const float4*)(gp + 4 * v);
      float4 b = *(const float4*)(gt + 4 * v);
      *(float4*)(&sP[4 * v]) = a;
      *(float4*)(&sT[4 * v]) = b;
    }
    __syncthreads();
#endif
    // lane t reads floats [t*30, t*30+30): stride 30 dwords, gcd(30,64)=2
    // -> 32 lanes hit 32 distinct banks, conflict-free.
    cell = cell_loss(&sP[tid * CH], &sT[tid * CH]);
  } else {
    int c = base + tid;
    if (c < n_cells)
      cell = cell_loss(pred + (size_t)c * CH, tgt + (size_t)c * CH);
  }

  // Deterministic block reduction: wave32 shuffle tree + LDS cross-wave.
  float v = wave_sum(cell);
  const int wid = tid >> 5, lane = tid & 31;
  if (lane == 0) sW[wid] = v;
  __syncthreads();
  if (wid == 0) {
    float w = (lane < NWAVES) ? sW[lane] : 0.0f;
#pragma unroll
    for (int off = NWAVES / 2; off > 0; off >>= 1)
      w += __shfl_down(w, off, 32);
    if (lane == 0) partial[blockIdx.x] = w;
  }
}

__global__ __launch_bounds__(TPB) void yolo_loss_final(
    const float* __restrict__ partial, float* __restrict__ out,
    int n, float scale) {
  __shared__ float sW[NWAVES];
  const int tid = threadIdx.x;
  float acc = 0.0f;
  for (int i = tid; i < n; i += TPB) acc += partial[i];   // fixed order
  float v = wave_sum(acc);
  const int wid = tid >> 5, lane = tid & 31;
  if (lane == 0) sW[wid] = v;
  __syncthreads();
  if (wid == 0) {
    float w = (lane < NWAVES) ? sW[lane] : 0.0f;
#pragma unroll
    for (int off = NWAVES / 2; off > 0; off >>= 1)
      w += __shfl_down(w, off, 32);
    if (lane == 0) out[0] = w * scale;
  }
}

extern "C" void kernel_launch(void* const* d_in, const int* in_sizes, int n_in,
                              void* d_out, int out_size, void* d_ws, size_t ws_size,
                              hipStream_t stream) {
  const float* pred = (const float*)d_in[0];
  const float* tgt  = (const float*)d_in[1];
  float* out = (float*)d_out;
  float* ws  = (float*)d_ws;

  const int n_cells = in_sizes[0] / CH;            // bs * S * S
  const int bs      = n_cells / (SS * SS);
  const int nblocks = (n_cells + TPB - 1) / TPB;

  yolo_loss_partial<<<nblocks, TPB, 0, stream>>>(pred, tgt, ws, n_cells);
  yolo_loss_final<<<1, TPB, 0, stream>>>(ws, out, nblocks, 1.0f / (float)bs);
}